// NoisyLinear_10840497455215
// MI455X (gfx1250) — compile-verified
//
#include <hip/hip_runtime.h>

typedef float v2f __attribute__((ext_vector_type(2)));
typedef float v4f __attribute__((ext_vector_type(4)));
typedef float v8f __attribute__((ext_vector_type(8)));

#define B_ 256
#define O_ 512
#define I_ 1024

// ---------------------------------------------------------------------------
// Kernel A: out[b,o] = sum_i x[b,i]*wmu[o,i] + bmu[o] + bsig[o]*beps[b,o]
// One wave per 16x16 output tile, V_WMMA_F32_16X16X4_F32 over K=1024.
//
// f32 WMMA operand layouts (ISA 7.12.2):
//   A (16x4, MxK): lanes 0-15 -> M=lane, v0=K0,v1=K1 ; lanes 16-31 -> M=lane-16, v0=K2,v1=K3
//   B (4x16, KxN): lanes 0-15 -> N=lane, v0=K0,v1=K1 ; lanes 16-31 -> N=lane-16, v0=K2,v1=K3
//   C/D (16x16):  VGPR r: lanes 0-15 -> (M=r, N=lane); lanes 16-31 -> (M=r+8, N=lane-16)
// ---------------------------------------------------------------------------
__global__ __launch_bounds__(256) void noisy_mu_gemm_wmma(
    const float* __restrict__ x,     // [B,I]
    const float* __restrict__ wmu,   // [O,I]
    const float* __restrict__ bmu,   // [O]
    const float* __restrict__ bsig,  // [O]
    const float* __restrict__ beps,  // [B,O]
    float* __restrict__ out)         // [B,O]
{
    const int lane = threadIdx.x & 31;
    const int wave = threadIdx.x >> 5;
    const int tile = blockIdx.x * 8 + wave;   // 0..511
    const int tn   = tile & 31;               // 32 tiles along O
    const int tm   = tile >> 5;               // 16 tiles along B
    const int half = lane >> 4;               // 0: K={0,1}, 1: K={2,3}
    const int lr   = lane & 15;

    const float* arow = x   + (size_t)(tm * 16 + lr) * I_;  // A: M = lr
    const float* brow = wmu + (size_t)(tn * 16 + lr) * I_;  // B: N = lr (wmu row o = B column)

    v8f c = {};
    #pragma unroll 4
    for (int k = 0; k < I_; k += 4) {
        v2f a = *(const v2f*)(arow + k + 2 * half);   // 8B aligned global_load_b64
        v2f b = *(const v2f*)(brow + k + 2 * half);
        c = __builtin_amdgcn_wmma_f32_16x16x4_f32(
                /*neg_a=*/false, a, /*neg_b=*/false, b,
                /*c_mod=*/(short)0, c, /*reuse_a=*/false, /*reuse_b=*/false);
    }

    const int   col = tn * 16 + lr;
    const float bm  = bmu[col];
    const float bs  = bsig[col];
    #pragma unroll
    for (int r = 0; r < 8; ++r) {
        const int row = tm * 16 + r + 8 * half;
        const size_t idx = (size_t)row * O_ + col;
        out[idx] = c[r] + bm + bs * beps[idx];
    }
}

// ---------------------------------------------------------------------------
// Kernel B: out[b,o] += sum_i wsig[o,i] * weps[b,o,i] * x[b,i]
// One wave per (b,o) row; 8 x 16B coalesced loads per lane; weps streamed
// non-temporally (537 MB, single use) so sigma/x rows stay resident in L2.
// ---------------------------------------------------------------------------
__global__ __launch_bounds__(256) void noisy_sigma_stream(
    const float* __restrict__ wsig,  // [O,I]
    const float* __restrict__ weps,  // [B,O,I]
    const float* __restrict__ x,     // [B,I]
    float* __restrict__ out)         // [B,O]
{
    const int lane = threadIdx.x & 31;
    const int wave = threadIdx.x >> 5;
    const int gid  = blockIdx.x * 8 + wave;  // 0 .. B*O-1 ; gid = b*O + o
    const int b    = gid >> 9;               // / 512
    const int o    = gid & 511;

    const v4f* wrow = (const v4f*)(weps + (size_t)gid * I_);
    const v4f* srow = (const v4f*)(wsig + (size_t)o  * I_);
    const v4f* xrow = (const v4f*)(x    + (size_t)b  * I_);

    v4f acc4 = {0.f, 0.f, 0.f, 0.f};
    #pragma unroll
    for (int j = 0; j < I_ / 128; ++j) {
        const int cidx = j * 32 + lane;
        v4f w  = __builtin_nontemporal_load(&wrow[cidx]);  // stream, don't cache
        v4f s  = srow[cidx];                               // L2-resident reuse
        v4f xv = xrow[cidx];                               // L2-resident reuse
        acc4 += s * w * xv;
    }
    float acc = acc4.x + acc4.y + acc4.z + acc4.w;

    #pragma unroll
    for (int off = 16; off > 0; off >>= 1)
        acc += __shfl_xor(acc, off, 32);

    if (lane == 0)
        out[gid] += acc;   // race-free: this wave uniquely owns (b,o); ordered after Kernel A
}

extern "C" void kernel_launch(void* const* d_in, const int* in_sizes, int n_in,
                              void* d_out, int out_size, void* d_ws, size_t ws_size,
                              hipStream_t stream) {
    const float* x    = (const float*)d_in[0];  // [B,I]
    const float* wmu  = (const float*)d_in[1];  // [O,I]
    const float* bmu  = (const float*)d_in[2];  // [O]
    const float* wsig = (const float*)d_in[3];  // [O,I]
    const float* bsig = (const float*)d_in[4];  // [O]
    const float* weps = (const float*)d_in[5];  // [B,O,I]
    const float* beps = (const float*)d_in[6];  // [B,O]
    float* out = (float*)d_out;                 // [B,O]

    // Kernel A: 512 tiles / 8 waves per block = 64 blocks
    noisy_mu_gemm_wmma<<<64, 256, 0, stream>>>(x, wmu, bmu, bsig, beps, out);
    // Kernel B: B*O rows / 8 waves per block = 16384 blocks
    noisy_sigma_stream<<<(B_ * O_) / 8, 256, 0, stream>>>(wsig, weps, x, out);
}